// CompetencyGNN_47218870452270
// MI455X (gfx1250) — compile-verified
//
#include <hip/hip_runtime.h>

typedef __attribute__((ext_vector_type(2))) float v2f;
typedef __attribute__((ext_vector_type(8))) float v8f;

#define H_F   16
#define OUT_F 8
#define C_F   1000

// ---------------- degree ----------------
__global__ void k_init_deg(float* __restrict__ deg, int N) {
    int i = blockIdx.x * blockDim.x + threadIdx.x;
    if (i < N) deg[i] = 1.0f;   // self-loop baked in
}

__global__ void k_count_deg(const int* __restrict__ dst, float* __restrict__ deg, int E) {
    int e = blockIdx.x * blockDim.x + threadIdx.x;
    if (e < E) atomicAdd(&deg[dst[e]], 1.0f);
}

__global__ void k_rsqrt_deg(float* __restrict__ dinv, int N) {
    int i = blockIdx.x * blockDim.x + threadIdx.x;
    if (i < N) dinv[i] = rsqrtf(dinv[i]);
}

// ---------------- layer 1: h1 = x@W1 ; agg1 = h1*dinv^2 (self loop) ----------------
__global__ void k_l1_node(const float* __restrict__ x, const float* __restrict__ W1,
                          const float* __restrict__ dinv,
                          float* __restrict__ h1, float* __restrict__ agg1, int N) {
    int i = blockIdx.x * blockDim.x + threadIdx.x;
    if (i >= N) return;
    float x0 = x[2 * i], x1 = x[2 * i + 1];
    float di = dinv[i];
    float d2 = di * di;
    float4* h1v = (float4*)(h1 + (size_t)i * H_F);
    float4* a1v = (float4*)(agg1 + (size_t)i * H_F);
#pragma unroll
    for (int q = 0; q < 4; ++q) {
        float4 hv;
        hv.x = x0 * W1[q * 4 + 0] + x1 * W1[H_F + q * 4 + 0];
        hv.y = x0 * W1[q * 4 + 1] + x1 * W1[H_F + q * 4 + 1];
        hv.z = x0 * W1[q * 4 + 2] + x1 * W1[H_F + q * 4 + 2];
        hv.w = x0 * W1[q * 4 + 3] + x1 * W1[H_F + q * 4 + 3];
        h1v[q] = hv;
        float4 av = {hv.x * d2, hv.y * d2, hv.z * d2, hv.w * d2};
        a1v[q] = av;
    }
}

__global__ void k_l1_edge(const int* __restrict__ src, const int* __restrict__ dst,
                          const float* __restrict__ dinv,
                          const float* __restrict__ h1, float* __restrict__ agg1, int E) {
    int e = blockIdx.x * blockDim.x + threadIdx.x;
    if (e >= E) return;
    int s = src[e], d = dst[e];
    float n = dinv[s] * dinv[d];
    const float4* hs = (const float4*)(h1 + (size_t)s * H_F);
    float* ad = agg1 + (size_t)d * H_F;
#pragma unroll
    for (int q = 0; q < 4; ++q) {
        float4 v = hs[q];
        atomicAdd(ad + q * 4 + 0, n * v.x);
        atomicAdd(ad + q * 4 + 1, n * v.y);
        atomicAdd(ad + q * 4 + 2, n * v.z);
        atomicAdd(ad + q * 4 + 3, n * v.w);
    }
}

// ---------------- layer 2: v = relu(agg1+b1); h2 = v@W2 ; agg2 = h2*dinv^2 ----------------
__global__ void k_l2_node(const float* __restrict__ agg1, const float* __restrict__ b1,
                          const float* __restrict__ W2, const float* __restrict__ dinv,
                          float* __restrict__ h2, float* __restrict__ agg2, int N) {
    int i = blockIdx.x * blockDim.x + threadIdx.x;
    if (i >= N) return;
    float v[H_F];
    const float4* ap = (const float4*)(agg1 + (size_t)i * H_F);
#pragma unroll
    for (int q = 0; q < 4; ++q) {
        float4 t = ap[q];
        float r0 = t.x + b1[q * 4 + 0]; v[q * 4 + 0] = r0 > 0.0f ? r0 : 0.0f;
        float r1 = t.y + b1[q * 4 + 1]; v[q * 4 + 1] = r1 > 0.0f ? r1 : 0.0f;
        float r2 = t.z + b1[q * 4 + 2]; v[q * 4 + 2] = r2 > 0.0f ? r2 : 0.0f;
        float r3 = t.w + b1[q * 4 + 3]; v[q * 4 + 3] = r3 > 0.0f ? r3 : 0.0f;
    }
    float o[OUT_F];
#pragma unroll
    for (int j = 0; j < OUT_F; ++j) o[j] = 0.0f;
#pragma unroll
    for (int f = 0; f < H_F; ++f) {
        float vf = v[f];
#pragma unroll
        for (int j = 0; j < OUT_F; ++j) o[j] += vf * W2[f * OUT_F + j];
    }
    float di = dinv[i];
    float d2 = di * di;
    float4* h2v = (float4*)(h2 + (size_t)i * OUT_F);
    float4* a2v = (float4*)(agg2 + (size_t)i * OUT_F);
    float4 p0 = {o[0], o[1], o[2], o[3]};
    float4 p1 = {o[4], o[5], o[6], o[7]};
    h2v[0] = p0; h2v[1] = p1;
    float4 q0 = {o[0] * d2, o[1] * d2, o[2] * d2, o[3] * d2};
    float4 q1 = {o[4] * d2, o[5] * d2, o[6] * d2, o[7] * d2};
    a2v[0] = q0; a2v[1] = q1;
}

__global__ void k_l2_edge(const int* __restrict__ src, const int* __restrict__ dst,
                          const float* __restrict__ dinv,
                          const float* __restrict__ h2, float* __restrict__ agg2, int E) {
    int e = blockIdx.x * blockDim.x + threadIdx.x;
    if (e >= E) return;
    int s = src[e], d = dst[e];
    float n = dinv[s] * dinv[d];
    const float4* hs = (const float4*)(h2 + (size_t)s * OUT_F);
    float* ad = agg2 + (size_t)d * OUT_F;
#pragma unroll
    for (int q = 0; q < 2; ++q) {
        float4 v = hs[q];
        atomicAdd(ad + q * 4 + 0, n * v.x);
        atomicAdd(ad + q * 4 + 1, n * v.y);
        atomicAdd(ad + q * 4 + 2, n * v.z);
        atomicAdd(ad + q * 4 + 3, n * v.w);
    }
}

// ---------------- final FC: out = (agg2 + b2) @ Wfc + bfc via V_WMMA_F32_16X16X4_F32 ----------------
// Handles ONLY full 16-row tiles (mtiles = N/16); leftover rows go to
// k_fc_rows_tail.  One wave owns a 16-row stripe; the A fragment (K=8 ->
// two K=4 slices) stays in registers while the wave sweeps the 62 full
// column tiles of Wfc (L2/L0-resident) with ZERO predication/clamping in
// the loop, then one predicated tail-column tile (cols 992..999).
// The kernel is bound by the 400 MB output store (~17 us at 23.3 TB/s).
__global__ void k_fc_wmma(const float* __restrict__ agg2, const float* __restrict__ b2,
                          const float* __restrict__ Wfc, const float* __restrict__ bfc,
                          float* __restrict__ out, int mtiles) {
    const int lane = threadIdx.x & 31;
    const int wave = threadIdx.x >> 5;
    const int mtile = blockIdx.x * 8 + wave;
    if (mtile >= mtiles) return;

    const int row0 = mtile * 16;
    const int half = lane >> 4;      // 0: lanes 0-15, 1: lanes 16-31
    const int m = lane & 15;

    // A fragment: 16x4 f32 layout -> lanes 0-15 carry K=0,1; lanes 16-31 K=2,3.
    // Second K-slice adds 4.  Bias b2 folded in.
    const int r = row0 + m;
    const float4* ap = (const float4*)(agg2 + (size_t)r * OUT_F);
    float4 t0 = ap[0], t1 = ap[1];
    float ar[8];
    ar[0] = t0.x + b2[0]; ar[1] = t0.y + b2[1]; ar[2] = t0.z + b2[2]; ar[3] = t0.w + b2[3];
    ar[4] = t1.x + b2[4]; ar[5] = t1.y + b2[5]; ar[6] = t1.z + b2[6]; ar[7] = t1.w + b2[7];
    const int kb0 = 2 * half;
    v2f a0 = {ar[kb0], ar[kb0 + 1]};
    v2f a1 = {ar[4 + kb0], ar[4 + kb0 + 1]};

    // per-lane output base: lanes 0-15 write rows row0..row0+7, lanes 16-31 rows row0+8..row0+15
    float* __restrict__ obase = out + (size_t)(row0 + 8 * half) * C_F + m;

    const int FULL_NT = C_F / 16;   // 62 full tiles, cols 0..991
    for (int nt = 0; nt < FULL_NT; ++nt) {
        const int col = nt * 16 + m;
        // B fragment: 4x16 f32 layout -> V0 = row kb0, V1 = row kb0+1 (per lane-half)
        v2f b0 = {Wfc[(size_t)(kb0 + 0) * C_F + col], Wfc[(size_t)(kb0 + 1) * C_F + col]};
        v2f b1 = {Wfc[(size_t)(kb0 + 4) * C_F + col], Wfc[(size_t)(kb0 + 5) * C_F + col]};
        v8f c = {};
        c = __builtin_amdgcn_wmma_f32_16x16x4_f32(false, a0, false, b0, (short)0, c, false, false);
        c = __builtin_amdgcn_wmma_f32_16x16x4_f32(false, a1, false, b1, (short)0, c, false, false);
        const float bias = bfc[col];
        float* po = obase + nt * 16;
#pragma unroll
        for (int v = 0; v < 8; ++v) po[(size_t)v * C_F] = c[v] + bias;   // C/D: VGPR v = row v (+8 upper half)
    }

    // tail-column tile: cols 992..1007, only 992..999 valid
    {
        const int col = FULL_NT * 16 + m;
        const int cc = col < C_F ? col : C_F - 1;
        v2f b0 = {Wfc[(size_t)(kb0 + 0) * C_F + cc], Wfc[(size_t)(kb0 + 1) * C_F + cc]};
        v2f b1 = {Wfc[(size_t)(kb0 + 4) * C_F + cc], Wfc[(size_t)(kb0 + 5) * C_F + cc]};
        v8f c = {};
        c = __builtin_amdgcn_wmma_f32_16x16x4_f32(false, a0, false, b0, (short)0, c, false, false);
        c = __builtin_amdgcn_wmma_f32_16x16x4_f32(false, a1, false, b1, (short)0, c, false, false);
        const float bias = bfc[cc];
        if (col < C_F) {
            float* po = obase + FULL_NT * 16;
#pragma unroll
            for (int v = 0; v < 8; ++v) po[(size_t)v * C_F] = c[v] + bias;
        }
    }
}

// Scalar fallback for the N % 16 leftover rows (zero rows for N=100000).
__global__ void k_fc_rows_tail(const float* __restrict__ agg2, const float* __restrict__ b2,
                               const float* __restrict__ Wfc, const float* __restrict__ bfc,
                               float* __restrict__ out, int row_start, int N) {
    int t = blockIdx.x * blockDim.x + threadIdx.x;
    int rows = N - row_start;
    if (t >= rows * C_F) return;
    int r = row_start + t / C_F;
    int c = t % C_F;
    float acc = bfc[c];
#pragma unroll
    for (int k = 0; k < OUT_F; ++k)
        acc += (agg2[(size_t)r * OUT_F + k] + b2[k]) * Wfc[(size_t)k * C_F + c];
    out[(size_t)r * C_F + c] = acc;
}

extern "C" void kernel_launch(void* const* d_in, const int* in_sizes, int n_in,
                              void* d_out, int out_size, void* d_ws, size_t ws_size,
                              hipStream_t stream) {
    const float* x   = (const float*)d_in[0];
    const int*   ei  = (const int*)d_in[1];
    const float* W1  = (const float*)d_in[2];
    const float* b1  = (const float*)d_in[3];
    const float* W2  = (const float*)d_in[4];
    const float* b2  = (const float*)d_in[5];
    const float* Wfc = (const float*)d_in[6];
    const float* bfc = (const float*)d_in[7];
    float* out = (float*)d_out;

    const int N = in_sizes[0] / 2;   // x is [N, 2]
    const int E = in_sizes[1] / 2;   // edge_index is [2, E]
    const int* src = ei;
    const int* dst = ei + E;

    // workspace layout (floats): dinv[N] | h1[16N] | agg1[16N] | h2[8N] | agg2[8N] = 49N ~ 19.6 MB
    float* ws   = (float*)d_ws;
    float* dinv = ws;
    float* h1   = ws + (size_t)N;
    float* agg1 = ws + (size_t)17 * N;
    float* h2   = ws + (size_t)33 * N;
    float* agg2 = ws + (size_t)41 * N;

    const int TB = 256;
    const int nb = (N + TB - 1) / TB;
    const int eb = (E + TB - 1) / TB;

    k_init_deg<<<nb, TB, 0, stream>>>(dinv, N);
    k_count_deg<<<eb, TB, 0, stream>>>(dst, dinv, E);
    k_rsqrt_deg<<<nb, TB, 0, stream>>>(dinv, N);

    k_l1_node<<<nb, TB, 0, stream>>>(x, W1, dinv, h1, agg1, N);
    k_l1_edge<<<eb, TB, 0, stream>>>(src, dst, dinv, h1, agg1, E);

    k_l2_node<<<nb, TB, 0, stream>>>(agg1, b1, W2, dinv, h2, agg2, N);
    k_l2_edge<<<eb, TB, 0, stream>>>(src, dst, dinv, h2, agg2, E);

    const int mtiles = N / 16;                  // full 16-row tiles only
    if (mtiles > 0) {
        const int gb = (mtiles + 7) / 8;        // 8 waves/block, 1 wave per 16-row stripe
        k_fc_wmma<<<gb, TB, 0, stream>>>(agg2, b2, Wfc, bfc, out, mtiles);
    }
    const int row_start = mtiles * 16;
    const int rem_rows = N - row_start;         // 0 for N=100000 (launch decision is N-only -> deterministic)
    if (rem_rows > 0) {
        const int tot = rem_rows * C_F;
        k_fc_rows_tail<<<(tot + TB - 1) / TB, TB, 0, stream>>>(agg2, b2, Wfc, bfc, out, row_start, N);
    }
}